// BatchedGAT_33036888441485
// MI455X (gfx1250) — compile-verified
//
#include <hip/hip_runtime.h>
#include <hip/hip_bf16.h>

#define BB    4
#define NN    512
#define IND   128
#define OUTD  128
#define NHEAD 4
#define PH    32
#define SLOPE 0.2f

typedef float v2f __attribute__((ext_vector_type(2)));
typedef float v8f __attribute__((ext_vector_type(8)));

// Optional CDNA5 async LDS-DMA path (guarded: falls back to sync staging).
#if defined(__has_builtin)
#if __has_builtin(__builtin_amdgcn_global_load_async_to_lds_b128) && \
    __has_builtin(__builtin_amdgcn_s_wait_asynccnt)
#define USE_ASYNC_LDS 1
#endif
#endif

#if defined(USE_ASYNC_LDS)
// Builtin signature (from hipcc diagnostic): first arg is
// 'int __vector(4) __device__ *'  == vector_size(16) int, address space 1.
typedef int v4i_vec __attribute__((vector_size(16)));
typedef __attribute__((address_space(1))) v4i_vec* gptr_v4i;
typedef __attribute__((address_space(3))) v4i_vec* lptr_v4i;
#endif

// ---------------------------------------------------------------------------
// wave32 shuffle reductions
// ---------------------------------------------------------------------------
__device__ __forceinline__ float wave_max32(float v) {
#pragma unroll
  for (int m = 16; m >= 1; m >>= 1) v = fmaxf(v, __shfl_xor(v, m, 32));
  return v;
}
__device__ __forceinline__ float wave_sum32(float v) {
#pragma unroll
  for (int m = 16; m >= 1; m >>= 1) v += __shfl_xor(v, m, 32);
  return v;
}

// ---------------------------------------------------------------------------
// Kernel 1: out[r, c] = sum_k x[r,k] * W[k,c] + bias[c]
// One wave per 16x16 tile, fp32 WMMA 16x16x4, K = 128.
// ---------------------------------------------------------------------------
__global__ void gat_linear_wmma(const float* __restrict__ x,
                                const float* __restrict__ W,
                                const float* __restrict__ bias,
                                float* __restrict__ out) {
  const int lane = threadIdx.x;
  const int ln   = lane & 15;
  const int half = lane >> 4;
  const int rowBase = blockIdx.x * 16;   // over B*N = 2048 rows
  const int colBase = blockIdx.y * 16;   // over OUTD = 128 cols

  v8f acc = {};
  const float* arow = x + (size_t)(rowBase + ln) * IND;
#pragma unroll 4
  for (int kt = 0; kt < IND / 4; ++kt) {
    const int k0 = kt * 4 + half * 2;
    v2f a = *(const v2f*)(arow + k0);
    v2f bv;
    bv.x = W[(size_t)(k0 + 0) * OUTD + colBase + ln];
    bv.y = W[(size_t)(k0 + 1) * OUTD + colBase + ln];
    acc = __builtin_amdgcn_wmma_f32_16x16x4_f32(false, a, false, bv,
                                                (short)0, acc, false, false);
  }
  const float bc = bias[colBase + ln];
#pragma unroll
  for (int v = 0; v < 8; ++v) {
    const int row = rowBase + v + 8 * half;
    out[(size_t)row * OUTD + colBase + ln] = acc[v] + bc;
  }
}

// ---------------------------------------------------------------------------
// Kernel 2: per (b, i): e[j,h] = att_h . leakyrelu(xr[b,i,h,:] + xl[b,j,h,:])
// masked (adj[b,j,i] != 0 || j==i), softmax over j, writes alpha[b,h,i,j].
// Block = 512 threads (one per source j) = 16 waves.
// Reduction: per-wave shfl_xor, then 16-entry cross-wave LDS combine.
// ---------------------------------------------------------------------------
__global__ __launch_bounds__(NN) void gat_scores_softmax(
    const float* __restrict__ xl, const float* __restrict__ xr,
    const int* __restrict__ adj, const float* __restrict__ att,
    float* __restrict__ alpha) {
  const int i = blockIdx.x;
  const int b = blockIdx.y;
  const int j = threadIdx.x;     // 0..511
  const int wave = j >> 5;       // 0..15
  const int lane = j & 31;

  __shared__ float xr_s[OUTD];
  __shared__ float att_s[OUTD];
  __shared__ float wred[16 * NHEAD];

  if (j < OUTD) {
    xr_s[j]  = xr[((size_t)b * NN + i) * OUTD + j];
    att_s[j] = att[j];           // att is [H, PH] flattened = 128 floats
  }
  __syncthreads();

  // per-thread scores for its source node j (float4 = global_load_b128)
  const float4* xlj4 = (const float4*)(xl + ((size_t)b * NN + j) * OUTD);
  float e[NHEAD];
#pragma unroll
  for (int h = 0; h < NHEAD; ++h) {
    float s = 0.f;
#pragma unroll
    for (int q = 0; q < PH / 4; ++q) {
      const float4 v4 = xlj4[h * (PH / 4) + q];
      const int c0 = h * PH + q * 4;
      float t;
      t = xr_s[c0 + 0] + v4.x; t = (t > 0.f) ? t : SLOPE * t; s = fmaf(att_s[c0 + 0], t, s);
      t = xr_s[c0 + 1] + v4.y; t = (t > 0.f) ? t : SLOPE * t; s = fmaf(att_s[c0 + 1], t, s);
      t = xr_s[c0 + 2] + v4.z; t = (t > 0.f) ? t : SLOPE * t; s = fmaf(att_s[c0 + 2], t, s);
      t = xr_s[c0 + 3] + v4.w; t = (t > 0.f) ? t : SLOPE * t; s = fmaf(att_s[c0 + 3], t, s);
    }
    e[h] = s;
  }
  const bool valid = (adj[((size_t)b * NN + j) * NN + i] != 0) || (j == i);

  // block max per head
#pragma unroll
  for (int h = 0; h < NHEAD; ++h) {
    const float v = wave_max32(valid ? e[h] : -3.0e38f);
    if (lane == 0) wred[wave * NHEAD + h] = v;
  }
  __syncthreads();
  float mx[NHEAD];
#pragma unroll
  for (int h = 0; h < NHEAD; ++h) {
    float v = -3.0e38f;
#pragma unroll
    for (int w = 0; w < 16; ++w) v = fmaxf(v, wred[w * NHEAD + h]);
    mx[h] = v;
  }
  __syncthreads();   // before wred reuse

  // exp + block sum per head
  float p[NHEAD];
#pragma unroll
  for (int h = 0; h < NHEAD; ++h) {
    p[h] = valid ? __expf(e[h] - mx[h]) : 0.f;
    const float s = wave_sum32(p[h]);
    if (lane == 0) wred[wave * NHEAD + h] = s;
  }
  __syncthreads();
#pragma unroll
  for (int h = 0; h < NHEAD; ++h) {
    float s = 0.f;
#pragma unroll
    for (int w = 0; w < 16; ++w) s += wred[w * NHEAD + h];
    alpha[(((size_t)b * NHEAD + h) * NN + i) * NN + j] = p[h] / s;
  }
}

// ---------------------------------------------------------------------------
// Kernel 3: out[b,i,h*32+c] = sum_j alpha[b,h,i,j] * xl[b,j,h*32+c] + bias
// Block = 128 threads (4 waves); the [512 x 16] B slice is staged once in LDS
// (32 KB, async LDS-DMA when available) and shared by 4 i-tiles; each wave
// runs a K=512 fp32 WMMA 16x16x4 chain.  B-read pattern from LDS is
// bank-conflict free (rows are 16 dwords).
// ---------------------------------------------------------------------------
__global__ __launch_bounds__(128) void gat_aggregate_wmma(
    const float* __restrict__ alpha, const float* __restrict__ xl,
    const float* __restrict__ bias, float* __restrict__ out) {
  __shared__ float Bs[NN][16];   // 32 KB

  const int tid  = threadIdx.x;  // 0..127
  const int wave = tid >> 5;     // 0..3
  const int lane = tid & 31;
  const int ln   = lane & 15;
  const int half = lane >> 4;
  const int cBase = blockIdx.y * 16;          // {0, 16}
  const int b = blockIdx.z >> 2;
  const int h = blockIdx.z & 3;
  const int iBase = blockIdx.x * 64 + wave * 16;

  // ---- stage B[k][n] = xl[b, k, h*PH + cBase + n] into LDS -------------
  const float* src = xl + (size_t)b * NN * OUTD + h * PH + cBase;
  const int colq = (tid & 3) * 4;   // float4 column chunk: 0,4,8,12
  const int row0 = tid >> 2;        // 0..31
#if defined(USE_ASYNC_LDS)
#pragma unroll
  for (int r = 0; r < 16; ++r) {
    const int k = row0 + r * 32;
    __builtin_amdgcn_global_load_async_to_lds_b128(
        (gptr_v4i)(src + (size_t)k * OUTD + colq),
        (lptr_v4i)&Bs[k][colq], 0, 0);
  }
  __builtin_amdgcn_s_wait_asynccnt(0);
  __syncthreads();
#else
#pragma unroll
  for (int r = 0; r < 16; ++r) {
    const int k = row0 + r * 32;
    *(float4*)&Bs[k][colq] = *(const float4*)(src + (size_t)k * OUTD + colq);
  }
  __syncthreads();
#endif

  // ---- WMMA chain over K = 512 -----------------------------------------
  const float* arow = alpha + (size_t)(b * NHEAD + h) * NN * NN +
                      (size_t)(iBase + ln) * NN;
  v8f acc = {};
#pragma unroll 2
  for (int kt = 0; kt < NN / 4; ++kt) {
    const int k0 = kt * 4 + half * 2;
    v2f a = *(const v2f*)(arow + k0);
    v2f bv;
    bv.x = Bs[k0 + 0][ln];
    bv.y = Bs[k0 + 1][ln];
    acc = __builtin_amdgcn_wmma_f32_16x16x4_f32(false, a, false, bv,
                                                (short)0, acc, false, false);
  }
  const float bc = bias[h * PH + cBase + ln];
#pragma unroll
  for (int v = 0; v < 8; ++v) {
    const int i = iBase + v + 8 * half;
    out[(size_t)b * NN * OUTD + (size_t)i * OUTD + h * PH + cBase + ln] =
        acc[v] + bc;
  }
}

// ---------------------------------------------------------------------------
extern "C" void kernel_launch(void* const* d_in, const int* in_sizes, int n_in,
                              void* d_out, int out_size, void* d_ws, size_t ws_size,
                              hipStream_t stream) {
  const float* x    = (const float*)d_in[0];
  const int*   adj  = (const int*)  d_in[1];
  const float* Wl   = (const float*)d_in[2];
  const float* bl   = (const float*)d_in[3];
  const float* Wr   = (const float*)d_in[4];
  const float* br   = (const float*)d_in[5];
  const float* att  = (const float*)d_in[6];
  const float* bias = (const float*)d_in[7];
  float* out = (float*)d_out;

  // Workspace layout (floats): XL | XR | ALPHA  (total ~18 MiB)
  float* ws    = (float*)d_ws;
  float* XL    = ws;                              // B*N*OUTD = 262144
  float* XR    = XL + (size_t)BB * NN * OUTD;     // 262144
  float* ALPHA = XR + (size_t)BB * NN * OUTD;     // B*H*N*N = 4194304

  // 1) Feature transforms (fp32 WMMA GEMMs)
  dim3 gLin((BB * NN) / 16, OUTD / 16);
  gat_linear_wmma<<<gLin, 32, 0, stream>>>(x, Wl, bl, XL);
  gat_linear_wmma<<<gLin, 32, 0, stream>>>(x, Wr, br, XR);

  // 2) Scores + masked softmax (VALU + shfl reductions) -> alpha[b,h,i,j]
  gat_scores_softmax<<<dim3(NN, BB), NN, 0, stream>>>(XL, XR, adj, att, ALPHA);

  // 3) Aggregation alpha @ xl + bias (fp32 WMMA, LDS-staged B tile)
  gat_aggregate_wmma<<<dim3(NN / 64, PH / 16, BB * NHEAD), 128, 0, stream>>>(
      ALPHA, XL, bias, out);
}